// CustomModel_12953621365157
// MI455X (gfx1250) — compile-verified
//
#include <hip/hip_runtime.h>
#include <hip/hip_bf16.h>

typedef __attribute__((ext_vector_type(2))) float v2f;
typedef __attribute__((ext_vector_type(8))) float v8f;

#define N_IN   512
#define P      512
#define LAYERS 4
#define NEDGE  32768
#define BATCH  4096
#define COLS_PER_WAVE 8   // 8 x 16 = 128 batch columns per wave

// u-buffer layout: K-interleaved groups of 4 so a WMMA B-operand lane reads
// one aligned float2:  u[((node>>2)*BATCH + col)*4 + (node&3)]
__device__ __forceinline__ size_t u_index(int node, int col) {
    return ((size_t)(node >> 2) * BATCH + col) * 4 + (node & 3);
}

// ---------------------------------------------------------------------------
__global__ void zero_f32(float* __restrict__ p, int n) {
    int i = blockIdx.x * blockDim.x + threadIdx.x;
    if (i < n) p[i] = 0.0f;
}

// Scatter edge list into dense count matrix A[dst, src] (P x n_src).
__global__ void build_A(float* __restrict__ A,
                        const int* __restrict__ src,
                        const int* __restrict__ dst,
                        int n_src) {
    int e = blockIdx.x * blockDim.x + threadIdx.x;
    if (e < NEDGE) {
        atomicAdd(&A[(size_t)dst[e] * n_src + src[e]], 1.0f);
    }
}

// Seed u for the input nodes: u = (|x|==1 ? x : w*x), relaid out K-interleaved.
__global__ void seed_u(float* __restrict__ u, const float* __restrict__ x,
                       const float* __restrict__ w_ptr) {
    int i = blockIdx.x * blockDim.x + threadIdx.x;
    if (i < N_IN * BATCH) {
        int node = i / BATCH, col = i - node * BATCH;
        float v = x[i];
        u[u_index(node, col)] = (fabsf(v) == 1.0f) ? v : v * w_ptr[0];
    }
}

// Per-node activation dispatch (activations_dict order).
__device__ __forceinline__ float apply_act(float x, int id) {
    switch (id) {
        case 0: return x;                         // linear
        case 1: return x < 0.0f ? -1.0f : 1.0f;   // step
        case 2: return sinf(x);                   // sin
        case 3: return cosf(x);                   // cos
        case 4: return expf(-x * x);              // Gaussian
        case 5: return tanhf(x);                  // tanh
        case 6: return 1.0f / (1.0f + expf(-x));  // sigmoid
        case 7: return fabsf(x);                  // abs
        case 8: return -x;                        // invert
        case 9: return x > 0.0f ? x : 0.0f;       // ReLU
    }
    return x;
}

// ---------------------------------------------------------------------------
// summed = A[P,n_src] @ u[n_src,BATCH] via V_WMMA_F32_16X16X4_F32, then
// activation.  One wave: 16 rows x 128 cols (8 accumulators). The 4 waves of
// a block share one column group -> identical B stream (L2/WGP$ friendly).
//
// f32 16x16x4 layouts (ISA 7.12.2), hi = lane>=16, lo = lane&15:
//   A: lane -> row row0+lo, VGPR j -> K = k + 2*hi + j   (float2 load)
//   B: VGPR j -> K = k + 2*hi + j, col = lo              (float2 via u layout)
//   C/D: VGPR r -> row row0 + r + 8*hi, col lo
__global__ __launch_bounds__(128) void neat_layer(
    const float* __restrict__ A,        // [P, n_src] count matrix
    const float* __restrict__ u_vals,   // K-interleaved u, all nodes so far
    float*       __restrict__ u_out,    // same buffer (next layer's operand)
    float*       __restrict__ raw_out,  // d_out for last layer, else nullptr
    const int*   __restrict__ act_ids,  // [P]
    const float* __restrict__ w_ptr,
    int n_src, int out_node_base)
{
    const int lane = threadIdx.x & 31;
    const int wave = blockIdx.x * (blockDim.x >> 5) + (threadIdx.x >> 5);

    const int row_tile = wave & 31;                     // 0..31  (P/16)
    const int colg     = wave >> 5;                     // 0..31  (BATCH/128)
    const int row0     = row_tile * 16;
    const int col0     = colg * (16 * COLS_PER_WAVE);
    const int hi       = lane >> 4;
    const int lo       = lane & 15;

    v8f c[COLS_PER_WAVE] = {};

    const float* arow = A + (size_t)(row0 + lo) * n_src + 2 * hi;
    const int    k4n  = n_src >> 2;

    for (int k4 = 0; k4 < k4n; ++k4) {
        v2f a = *(const v2f*)(arow + 4 * k4);           // K = 4k4+2hi, +1
        const float* ub =
            u_vals + ((size_t)k4 * BATCH + col0 + lo) * 4 + 2 * hi;
#pragma unroll
        for (int t = 0; t < COLS_PER_WAVE; ++t) {
            v2f b = *(const v2f*)(ub + t * 64);         // 16 cols * 4 floats
            c[t] = __builtin_amdgcn_wmma_f32_16x16x4_f32(
                       false, a, false, b, (short)0, c[t], false, false);
        }
    }

    const float w = w_ptr[0];
#pragma unroll
    for (int r = 0; r < 8; ++r) {
        const int row = row0 + r + 8 * hi;
        const int id  = act_ids[row];
#pragma unroll
        for (int t = 0; t < COLS_PER_WAVE; ++t) {
            const int  col = col0 + t * 16 + lo;
            const float y  = apply_act(c[t][r], id);
            if (raw_out) {
                raw_out[(size_t)row * BATCH + col] = y;   // final layer output
            } else {
                // pre-apply the +/-1 quirk for the next layer's B operand
                const float uq = (fabsf(y) == 1.0f) ? y : y * w;
                u_out[u_index(out_node_base + row, col)] = uq;
            }
        }
    }
}

// ---------------------------------------------------------------------------
extern "C" void kernel_launch(void* const* d_in, const int* in_sizes, int n_in,
                              void* d_out, int out_size, void* d_ws, size_t ws_size,
                              hipStream_t stream) {
    const float* x        = (const float*)d_in[0];   // [N_IN, BATCH]
    const float* w        = (const float*)d_in[1];   // scalar
    const int*   edge_src = (const int*)d_in[2];     // [LAYERS, NEDGE]
    const int*   edge_dst = (const int*)d_in[3];     // [LAYERS, NEDGE]
    const int*   act_ids  = (const int*)d_in[4];     // [LAYERS, P]
    float*       out      = (float*)d_out;           // [P, BATCH]

    // Workspace: u_vals [2048 nodes x BATCH] f32 (32 MB) + A [P x 2048] (4 MB)
    float* u_vals = (float*)d_ws;
    float* A      = (float*)d_ws + (size_t)(N_IN + 3 * P) * BATCH;

    seed_u<<<(N_IN * BATCH + 255) / 256, 256, 0, stream>>>(u_vals, x, w);

    const int waves  = (P / 16) * (BATCH / (16 * COLS_PER_WAVE));  // 1024
    const int blocks = waves * 32 / 128;

    for (int l = 0; l < LAYERS; ++l) {
        const int n_src = N_IN + l * P;
        const int nA    = P * n_src;

        zero_f32<<<(nA + 255) / 256, 256, 0, stream>>>(A, nA);
        build_A<<<(NEDGE + 255) / 256, 256, 0, stream>>>(
            A, edge_src + (size_t)l * NEDGE, edge_dst + (size_t)l * NEDGE, n_src);

        const bool last = (l == LAYERS - 1);
        neat_layer<<<blocks, 128, 0, stream>>>(
            A, u_vals, u_vals, last ? out : nullptr,
            act_ids + (size_t)l * P, w, n_src, N_IN + l * P);
    }
}